// TextGCNLayer_76828374991033
// MI455X (gfx1250) — compile-verified
//
#include <hip/hip_runtime.h>

typedef __attribute__((ext_vector_type(2))) float v2f;
typedef __attribute__((ext_vector_type(8))) float v8f;

// Types matching the probed builtin signature:
//   __builtin_amdgcn_global_load_async_to_lds_b128(v4i global*, v4i lds*, imm, imm)
typedef __attribute__((__vector_size__(4 * sizeof(int)))) int v4i_;
typedef v4i_ __attribute__((address_space(1)))* gv4i_p;   // global int4*
typedef v4i_ __attribute__((address_space(3)))* lv4i_p;   // LDS    int4*

#define N_NODES  100000
#define IN_FEAT  256
#define OUT_FEAT 128
#define N_EDGES  3200000

// ---- gfx1250 async global->LDS copy (ASYNCcnt-tracked), with fallback ------
#if defined(__has_builtin)
#if __has_builtin(__builtin_amdgcn_global_load_async_to_lds_b128)
#define HAVE_ASYNC_LDS 1
#endif
#if __has_builtin(__builtin_amdgcn_s_wait_asynccnt)
#define HAVE_WAIT_ASYNC 1
#endif
#endif
#ifndef HAVE_ASYNC_LDS
#define HAVE_ASYNC_LDS 0
#endif
#ifndef HAVE_WAIT_ASYNC
#define HAVE_WAIT_ASYNC 0
#endif

// ---------------------------------------------------------------------------
// Phase 0: zero the output (we accumulate into it with atomics in phase 2).
// 12.8M floats = 3.2M float4 (exact), grid-stride b128 stores.
// ---------------------------------------------------------------------------
__global__ void zero_f32_kernel(float4* __restrict__ p, long n4) {
    long i = (long)blockIdx.x * blockDim.x + threadIdx.x;
    long stride = (long)gridDim.x * blockDim.x;
    const float4 z = make_float4(0.f, 0.f, 0.f, 0.f);
    for (; i < n4; i += stride) p[i] = z;
}

// ---------------------------------------------------------------------------
// Phase 1: support = X @ W using native fp32 WMMA (V_WMMA_F32_16X16X4_F32).
//
// Grid: 6250 blocks (one per 16-row M strip), 256 threads = 8 waves.
// The 16x256 fp32 strip of X (16 KB, contiguous) is staged into LDS ONCE per
// block via async global->LDS b128 copies (ASYNCcnt), removing the 8x
// redundant A traffic; each wave then feeds its A operand from LDS
// (ds_load_b64) while computing its own 16x16 N tile with 64 chained WMMAs.
//
// A 16x4 fp32 layout (ISA 7.12.2): lanes 0-15 hold row M=lane, K=k..k+1;
// lanes 16-31 hold row M=lane-16, K=k+2..k+3.
// B 4x16: VGPR0 = rows K=k (lanes 0-15) / K=k+1 (lanes 16-31); VGPR1 = K+2/K+3.
// C/D 16x16 f32: VGPR r -> M=r (lanes 0-15), M=8+r (lanes 16-31).
// ---------------------------------------------------------------------------
__global__ void gemm_xw_wmma_kernel(const float* __restrict__ X,
                                    const float* __restrict__ W,
                                    float* __restrict__ support) {
    __shared__ float As[16 * IN_FEAT];   // 16 KB A strip

    const int tid  = threadIdx.x;
    const int lane = tid & 31;
    const int wave = tid >> 5;           // 0..7 -> N tile index
    const int m0   = blockIdx.x * 16;
    const int n0   = wave * 16;
    const int half = lane >> 4;          // 0 for lanes 0-15, 1 for lanes 16-31
    const int l15  = lane & 15;

    // ---- stage the contiguous 16KB X strip into LDS ----
    const float* __restrict__ strip = X + (long)m0 * IN_FEAT;
#if HAVE_ASYNC_LDS
    #pragma unroll
    for (int i = 0; i < 4; ++i) {
        const int off = (tid + i * 256) * 4;  // dword index, 16B granules
        __builtin_amdgcn_global_load_async_to_lds_b128(
            (gv4i_p)(strip + off), (lv4i_p)(As + off), 0, 0);
    }
#if HAVE_WAIT_ASYNC
    __builtin_amdgcn_s_wait_asynccnt(0);
#else
    asm volatile("s_wait_asynccnt 0" ::: "memory");
#endif
#else
    #pragma unroll
    for (int i = 0; i < 4; ++i) {
        const int off = (tid + i * 256) * 4;
        *(float4*)(As + off) = *(const float4*)(strip + off);
    }
#endif
    __syncthreads();

    const float* arow = As + l15 * IN_FEAT + 2 * half;  // LDS source for A

    v8f c = {};
    #pragma unroll 8
    for (int k = 0; k < IN_FEAT; k += 4) {
        v2f a;                                           // ds_load_b64
        a.x = arow[k];                                   // K = k   + 2*half
        a.y = arow[k + 1];                               // K = k+1 + 2*half
        v2f b;
        b.x = W[(long)(k + half)     * OUT_FEAT + n0 + l15];  // K rows k / k+1
        b.y = W[(long)(k + 2 + half) * OUT_FEAT + n0 + l15];  // K rows k+2 / k+3
        // 8 args: (neg_a, A, neg_b, B, c_mod, C, reuse_a, reuse_b)
        c = __builtin_amdgcn_wmma_f32_16x16x4_f32(false, a, false, b,
                                                  (short)0, c, false, false);
    }

    float* __restrict__ crow =
        support + (long)(m0 + half * 8) * OUT_FEAT + n0 + l15;
    #pragma unroll
    for (int r = 0; r < 8; ++r) {
        crow[(long)r * OUT_FEAT] = c[r];
    }
}

// ---------------------------------------------------------------------------
// Phase 2: out[r] += vals[e] * support[cols[e]]  (segment-sum, rows sorted)
//
// One wave owns 128 contiguous edges, processed as 4 batches of 32:
// one coalesced load per edge array per batch, then __shfl broadcasts
// (ds_bpermute / v_readlane -- no memory traffic) feed the scalar edge data.
// Each lane owns 4 features (lane, +32, +64, +96) -> fully coalesced 512B
// gather of a support row, L2-resident (51.2 MB < 192 MB L2).
// Register-accumulate while the row id is unchanged; flush with
// global_atomic_add_f32 on row transitions / chunk boundaries.
// N_EDGES = 3.2M = 25000 waves * 128 edges exactly -> no tail handling.
// ---------------------------------------------------------------------------
#define EDGES_PER_WAVE 128

__global__ void spmm_segsum_kernel(const int*   __restrict__ rows,
                                   const int*   __restrict__ cols,
                                   const float* __restrict__ vals,
                                   const float* __restrict__ support,
                                   float*       __restrict__ out) {
    const int  lane = threadIdx.x & 31;
    const long wave = (long)blockIdx.x * (blockDim.x >> 5) + (threadIdx.x >> 5);
    const long e0   = wave * EDGES_PER_WAVE;

    float a0 = 0.f, a1 = 0.f, a2 = 0.f, a3 = 0.f;
    int cur = -1;

    for (int b = 0; b < EDGES_PER_WAVE / 32; ++b) {
        const long base = e0 + b * 32;
        const int   rb = rows[base + lane];   // coalesced batch loads
        const int   cb = cols[base + lane];
        const float vb = vals[base + lane];

        for (int i = 0; i < 32; ++i) {
            const int   r = __shfl(rb, i, 32);    // wave-uniform broadcasts
            const int   ci = __shfl(cb, i, 32);
            const float v = __shfl(vb, i, 32);

            if (r != cur) {                       // uniform branch
                if (cur >= 0) {
                    float* o = out + (long)cur * OUT_FEAT + lane;
                    atomicAdd(o,      a0);
                    atomicAdd(o + 32, a1);
                    atomicAdd(o + 64, a2);
                    atomicAdd(o + 96, a3);
                }
                cur = r;
                a0 = a1 = a2 = a3 = 0.f;
            }

            const float* s = support + (long)ci * OUT_FEAT + lane;
            a0 = fmaf(v, s[0],  a0);
            a1 = fmaf(v, s[32], a1);
            a2 = fmaf(v, s[64], a2);
            a3 = fmaf(v, s[96], a3);
        }
    }

    if (cur >= 0) {
        float* o = out + (long)cur * OUT_FEAT + lane;
        atomicAdd(o,      a0);
        atomicAdd(o + 32, a1);
        atomicAdd(o + 64, a2);
        atomicAdd(o + 96, a3);
    }
}

// ---------------------------------------------------------------------------
// Launch: zero out -> WMMA GEMM into d_ws -> segmented SpMM into d_out.
// ---------------------------------------------------------------------------
extern "C" void kernel_launch(void* const* d_in, const int* in_sizes, int n_in,
                              void* d_out, int out_size, void* d_ws, size_t ws_size,
                              hipStream_t stream) {
    const float* X    = (const float*)d_in[0];
    const float* W    = (const float*)d_in[1];
    const int*   rows = (const int*)  d_in[2];
    const int*   cols = (const int*)  d_in[3];
    const float* vals = (const float*)d_in[4];

    float* out     = (float*)d_out;
    float* support = (float*)d_ws;   // needs 100000*128*4 = 51.2 MB

    (void)in_sizes; (void)n_in; (void)out_size; (void)ws_size;

    // 1) out = 0  (3.2M float4, exact)
    zero_f32_kernel<<<2048, 256, 0, stream>>>((float4*)out,
                                              (long)N_NODES * OUT_FEAT / 4);

    // 2) support = X @ W   (6250 M-strips x 8 wave-tiles, A staged via LDS)
    gemm_xw_wmma_kernel<<<N_NODES / 16, 256, 0, stream>>>(X, W, support);

    // 3) out += A @ support  (25000 waves * 128 edges, exact)
    const int n_blocks = N_EDGES / (EDGES_PER_WAVE * 8);   // 3125
    spmm_segsum_kernel<<<n_blocks, 256, 0, stream>>>(rows, cols, vals,
                                                     support, out);
}